// GCN_8_plus2_6408091205939
// MI455X (gfx1250) — compile-verified
//
#include <hip/hip_runtime.h>
#include <math.h>

// ---------------- problem constants (match reference) ----------------
#define N_NODES   16384
#define E_EDGES   100000
#define FIN       128
#define NCLS      16
#define HEADS     8
#define EDIM      12
#define EF_EDGES  (E_EDGES + N_NODES)     // edges + self loops
#define EPS_F     1e-5f
#define NEG_SLOPE 0.2f

typedef __attribute__((ext_vector_type(2))) float v2f;
typedef __attribute__((ext_vector_type(8))) float v8f;

// ------------- ordered-float encoding for atomicMax on u32 -------------
__device__ __forceinline__ unsigned fenc(float f) {
  unsigned u = __float_as_uint(f);
  return (u & 0x80000000u) ? ~u : (u | 0x80000000u);
}
__device__ __forceinline__ float fdec(unsigned u) {
  return (u & 0x80000000u) ? __uint_as_float(u & 0x7FFFFFFFu) : __uint_as_float(~u);
}
#define ENC_NEG_INF 0x007FFFFFu   // fenc(-inf)

// ---------------------------- fills ----------------------------
__global__ void fill_f32(float* __restrict__ p, float v, long long n) {
  long long i = (long long)blockIdx.x * blockDim.x + threadIdx.x;
  long long s = (long long)gridDim.x * blockDim.x;
  for (; i < n; i += s) p[i] = v;
}
__global__ void fill_u32(unsigned* __restrict__ p, unsigned v, long long n) {
  long long i = (long long)blockIdx.x * blockDim.x + threadIdx.x;
  long long s = (long long)gridDim.x * blockDim.x;
  for (; i < n; i += s) p[i] = v;
}

// ------------- WMMA fp32 GEMM:  C[M,N] = A[M,K] @ W[N,K]^T (+bias) -------------
// One wave owns a 16x(16*NT) strip: the A fragment is loaded ONCE per K-step and
// reused across NT WMMAs (NT=4 -> 4x arithmetic intensity on the A stream).
// Fragment loads are software-pipelined one K-step ahead so s_wait_loadcnt
// overlaps the WMMA issue. 2 waves / block.
// V_WMMA_F32_16X16X4_F32 layout (ISA 7.12.2):
//   A 16x4 : lane l (r=l&15, hf=l>>4) holds A[r][2*hf+0..1]
//   B 4x16 : lane l holds B[2*hf+0..1][r]  (B = W^T -> contiguous in W rows)
//   C/D    : vgpr v, lane l -> row v+8*hf, col r
// Launch contract: M % 16 == 0 and N % (16*NT) == 0 (no guards needed).
template <int NT>
__global__ __launch_bounds__(64)
void wmma_gemm_nt(const float* __restrict__ A, const float* __restrict__ W,
                  const float* __restrict__ bias, float* __restrict__ C,
                  int M, int N, int K)
{
  int wave = threadIdx.x >> 5, lane = threadIdx.x & 31;
  int tileM = blockIdx.x * 16;
  int n0 = (blockIdx.y * 2 + wave) * (16 * NT);
  if (tileM >= M || n0 >= N) return;
  int hf = lane >> 4, r = lane & 15;
  const float* Arow = A + (size_t)(tileM + r) * K + 2 * hf;
  const float* Brow[NT];
#pragma unroll
  for (int t = 0; t < NT; ++t)
    Brow[t] = W + (size_t)(n0 + 16 * t + r) * K + 2 * hf;

  v8f acc[NT] = {};
  v2f a = *reinterpret_cast<const v2f*>(Arow);
  v2f b[NT];
#pragma unroll
  for (int t = 0; t < NT; ++t) b[t] = *reinterpret_cast<const v2f*>(Brow[t]);

  for (int k = 0; k < K; k += 4) {
    v2f an = a;
    v2f bn[NT];
#pragma unroll
    for (int t = 0; t < NT; ++t) bn[t] = b[t];
    if (k + 4 < K) {                       // pipeline next fragments
      an = *reinterpret_cast<const v2f*>(Arow + k + 4);
#pragma unroll
      for (int t = 0; t < NT; ++t)
        bn[t] = *reinterpret_cast<const v2f*>(Brow[t] + k + 4);
      if (k + 68 < K) {                    // near (WGP-scope) prefetch
        __builtin_prefetch(Arow + k + 68, 0, 3);
        __builtin_prefetch(Brow[0] + k + 68, 0, 3);
      }
    }
#pragma unroll
    for (int t = 0; t < NT; ++t)
      acc[t] = __builtin_amdgcn_wmma_f32_16x16x4_f32(false, a, false, b[t],
                                                     (short)0, acc[t], false, false);
    a = an;
#pragma unroll
    for (int t = 0; t < NT; ++t) b[t] = bn[t];
  }
#pragma unroll
  for (int t = 0; t < NT; ++t) {
    int col = n0 + 16 * t + r;
    float bv = bias ? bias[col] : 0.0f;
    float* Cp = C + (size_t)(tileM + 8 * hf) * N + col;
#pragma unroll
    for (int v = 0; v < 8; ++v) Cp[(size_t)v * N] = acc[t][v] + bv;
  }
}

// --------------- fused edge MLP: 12 -> 48 -> 24 -> 12 (weights in LDS) ---------------
__global__ __launch_bounds__(256)
void edge_mlp_kernel(const float* __restrict__ ea,
                     const float* __restrict__ w1, const float* __restrict__ b1,
                     const float* __restrict__ w2, const float* __restrict__ b2,
                     const float* __restrict__ w3, const float* __restrict__ b3,
                     float* __restrict__ e_out)
{
  __shared__ float sw1[48 * 12], sb1[48], sw2[24 * 48], sb2[24], sw3[12 * 24], sb3[12];
  for (int i = threadIdx.x; i < 48 * 12; i += 256) sw1[i] = w1[i];
  for (int i = threadIdx.x; i < 24 * 48; i += 256) sw2[i] = w2[i];
  for (int i = threadIdx.x; i < 12 * 24; i += 256) sw3[i] = w3[i];
  if (threadIdx.x < 48) sb1[threadIdx.x] = b1[threadIdx.x];
  if (threadIdx.x < 24) sb2[threadIdx.x] = b2[threadIdx.x];
  if (threadIdx.x < 12) sb3[threadIdx.x] = b3[threadIdx.x];
  __syncthreads();
  int e = blockIdx.x * 256 + threadIdx.x;
  if (e >= E_EDGES) return;
  float x[EDIM];
#pragma unroll
  for (int k = 0; k < EDIM; ++k) x[k] = ea[(size_t)e * EDIM + k];
  float h1[48];
#pragma unroll 4
  for (int o = 0; o < 48; ++o) {
    float s = sb1[o];
#pragma unroll
    for (int k = 0; k < 12; ++k) s += sw1[o * 12 + k] * x[k];
    h1[o] = fmaxf(s, 0.0f);
  }
  float h2[24];
#pragma unroll 4
  for (int o = 0; o < 24; ++o) {
    float s = sb2[o];
#pragma unroll 8
    for (int k = 0; k < 48; ++k) s += sw2[o * 48 + k] * h1[k];
    h2[o] = fmaxf(s, 0.0f);
  }
#pragma unroll
  for (int o = 0; o < 12; ++o) {
    float s = sb3[o];
#pragma unroll 8
    for (int k = 0; k < 24; ++k) s += sw3[o * 24 + k] * h2[k];
    e_out[(size_t)e * EDIM + o] = s;
  }
}

// --------------- self-loop edge features: fill_value='mean' ---------------
__global__ void edge_scatter_sum(const int* __restrict__ ei, const float* __restrict__ e,
                                 float* __restrict__ esum, float* __restrict__ cnt)
{
  int t = blockIdx.x * blockDim.x + threadIdx.x;
  if (t >= E_EDGES * EDIM) return;
  int i = t / EDIM, k = t - i * EDIM;
  int d = ei[E_EDGES + i];
  atomicAdd(&esum[(size_t)d * EDIM + k], e[t]);
  if (k == 0) atomicAdd(&cnt[d], 1.0f);
}
__global__ void selfloop_mean(const float* __restrict__ esum, const float* __restrict__ cnt,
                              float* __restrict__ e_f)
{
  int t = blockIdx.x * blockDim.x + threadIdx.x;
  if (t >= N_NODES * EDIM) return;
  int n = t / EDIM;
  e_f[(size_t)(E_EDGES + n) * EDIM + (t - n * EDIM)] = esum[t] / fmaxf(cnt[n], 1.0f);
}

// --------------- per-node attention scores (wave32 per (n,h), coalesced) ---------------
__global__ __launch_bounds__(256)
void att_node_kernel(const float* __restrict__ xp,
                     const float* __restrict__ att_src,
                     const float* __restrict__ att_dst,
                     float* __restrict__ a_s, float* __restrict__ a_d, int C)
{
  int wave = threadIdx.x >> 5, lane = threadIdx.x & 31;
  int t = blockIdx.x * 8 + wave;
  if (t >= N_NODES * HEADS) return;
  int n = t / HEADS, h = t - n * HEADS;
  const float* xr = xp + (size_t)n * HEADS * C + (size_t)h * C;
  const float* as = att_src + h * C;
  const float* ad = att_dst + h * C;
  float s = 0.f, d = 0.f;
  for (int c = lane; c < C; c += 32) { float v = xr[c]; s += v * as[c]; d += v * ad[c]; }
#pragma unroll
  for (int off = 16; off > 0; off >>= 1) {
    s += __shfl_xor(s, off, 32);
    d += __shfl_xor(d, off, 32);
  }
  if (lane == 0) { a_s[t] = s; a_d[t] = d; }
}

// a_e[i,h] = e_f[i] . ve[h]  with  ve[h,k] = sum_c We[h*C+c, k] * att_edge[h,c]
// (the edge projection ep is never materialized -> avoids a ~950MB intermediate)
__global__ void att_edge_vec_kernel(const float* __restrict__ We,
                                    const float* __restrict__ att_e,
                                    float* __restrict__ ve, int C)
{
  int t = blockIdx.x * blockDim.x + threadIdx.x;
  if (t >= HEADS * EDIM) return;
  int h = t / EDIM, k = t - h * EDIM;
  float s = 0.f;
  for (int c = 0; c < C; ++c) s += We[((size_t)(h * C + c)) * EDIM + k] * att_e[h * C + c];
  ve[t] = s;
}

// --------------- edge logits + per-dst running max ---------------
__global__ void logits_max_kernel(const int* __restrict__ ei, const float* __restrict__ e_f,
                                  const float* __restrict__ ve, const float* __restrict__ a_s,
                                  const float* __restrict__ a_d, float* __restrict__ logits,
                                  unsigned* __restrict__ mEnc)
{
  int t = blockIdx.x * blockDim.x + threadIdx.x;
  if (t >= EF_EDGES * HEADS) return;
  int i = t / HEADS, h = t - i * HEADS;
  int s = (i < E_EDGES) ? ei[i] : (i - E_EDGES);
  int d = (i < E_EDGES) ? ei[E_EDGES + i] : (i - E_EDGES);
  const float* er = e_f + (size_t)i * EDIM;
  const float* vh = ve + h * EDIM;
  float ae = 0.f;
#pragma unroll
  for (int k = 0; k < EDIM; ++k) ae += er[k] * vh[k];
  float l = a_s[s * HEADS + h] + a_d[d * HEADS + h] + ae;
  l = (l >= 0.f) ? l : NEG_SLOPE * l;
  logits[t] = l;
  atomicMax(&mEnc[d * HEADS + h], fenc(l));
}

// --------------- w = exp(l - m[dst]); den[dst] += w ---------------
__global__ void expw_kernel(const int* __restrict__ ei, float* __restrict__ logits,
                            const unsigned* __restrict__ mEnc, float* __restrict__ den)
{
  int t = blockIdx.x * blockDim.x + threadIdx.x;
  if (t >= EF_EDGES * HEADS) return;
  int i = t / HEADS, h = t - i * HEADS;
  int d = (i < E_EDGES) ? ei[E_EDGES + i] : (i - E_EDGES);
  float w = __expf(logits[t] - fdec(mEnc[d * HEADS + h]));
  logits[t] = w;
  atomicAdd(&den[d * HEADS + h], w);
}

// --------------- out[dst] starts at bias; += alpha * xp[src] ---------------
__global__ void agg_init_kernel(float* __restrict__ out, const float* __restrict__ bias, int HC)
{
  long long n = (long long)N_NODES * HC;
  long long stride = (long long)gridDim.x * blockDim.x;
  for (long long t = (long long)blockIdx.x * blockDim.x + threadIdx.x; t < n; t += stride)
    out[t] = bias[(int)(t % HC)];
}
__global__ void aggregate_kernel(const int* __restrict__ ei, const float* __restrict__ xp,
                                 const float* __restrict__ w, const float* __restrict__ den,
                                 float* __restrict__ out, int C)
{
  int HC = HEADS * C;
  long long total = (long long)EF_EDGES * HC;
  long long stride = (long long)gridDim.x * blockDim.x;
  for (long long t = (long long)blockIdx.x * blockDim.x + threadIdx.x; t < total; t += stride) {
    long long ii64 = t / HC;
    int j = (int)(t - ii64 * HC);
    int i = (int)ii64;
    int h = j / C;
    int s = (i < E_EDGES) ? ei[i] : (i - E_EDGES);
    int d = (i < E_EDGES) ? ei[E_EDGES + i] : (i - E_EDGES);
    float alpha = w[(size_t)i * HEADS + h] / den[d * HEADS + h];
    atomicAdd(&out[(size_t)d * HC + j], alpha * xp[(size_t)s * HC + j]);
  }
}

// --------------- BatchNorm (+ReLU, in-place) ---------------
__global__ __launch_bounds__(256)
void bn_partial_kernel(const float* __restrict__ x, float* __restrict__ stats, int Ccols)
{
  int col = blockIdx.x * 256 + threadIdx.x;      // gridDim.x*256 == Ccols
  int rows = N_NODES / gridDim.y;
  int r0 = blockIdx.y * rows;
  float s = 0.f, q = 0.f;
  for (int r = 0; r < rows; ++r) {
    float v = x[(size_t)(r0 + r) * Ccols + col];
    s += v; q += v * v;
  }
  atomicAdd(&stats[col], s);
  atomicAdd(&stats[Ccols + col], q);
}
__global__ void bn_apply_kernel(float* __restrict__ x, const float* __restrict__ stats,
                                const float* __restrict__ g, const float* __restrict__ b, int Ccols)
{
  long long n = (long long)N_NODES * Ccols;
  long long stride = (long long)gridDim.x * blockDim.x;
  const float invN = 1.0f / (float)N_NODES;
  for (long long t = (long long)blockIdx.x * blockDim.x + threadIdx.x; t < n; t += stride) {
    int col = (int)(t % Ccols);
    float mu = stats[col] * invN;
    float var = stats[Ccols + col] * invN - mu * mu;
    float y = (x[t] - mu) * rsqrtf(var + EPS_F) * g[col] + b[col];
    x[t] = fmaxf(y, 0.0f);
  }
}

// --------------- LayerNorm + ReLU + residual add (wave32 per row) ---------------
__global__ __launch_bounds__(256)
void ln_relu_res_kernel(const float* __restrict__ hl, const float* __restrict__ resv,
                        const float* __restrict__ g, const float* __restrict__ b,
                        float* __restrict__ out, int Dl)
{
  int wave = threadIdx.x >> 5, lane = threadIdx.x & 31;
  int row = blockIdx.x * 8 + wave;
  if (row >= N_NODES) return;
  const float* xr = hl + (size_t)row * Dl;
  int per = Dl >> 5;                 // 2, 4 or 8
  float vals[8];
  float s = 0.f;
  for (int k = 0; k < per; ++k) { float v = xr[lane + 32 * k]; vals[k] = v; s += v; }
#pragma unroll
  for (int off = 16; off > 0; off >>= 1) s += __shfl_xor(s, off, 32);
  float mu = s / (float)Dl;
  float q = 0.f;
  for (int k = 0; k < per; ++k) { float c = vals[k] - mu; q += c * c; }
#pragma unroll
  for (int off = 16; off > 0; off >>= 1) q += __shfl_xor(q, off, 32);
  float inv = rsqrtf(q / (float)Dl + EPS_F);
  float* orow = out + (size_t)row * Dl;
  const float* rrow = resv + (size_t)row * Dl;
  for (int k = 0; k < per; ++k) {
    int c = lane + 32 * k;
    float y = (vals[k] - mu) * inv * g[c] + b[c];
    orow[c] = fmaxf(y, 0.0f) + rrow[c];
  }
}

// ============================ host orchestration ============================
extern "C" void kernel_launch(void* const* d_in, const int* in_sizes, int n_in,
                              void* d_out, int out_size, void* d_ws, size_t ws_size,
                              hipStream_t stream)
{
  (void)in_sizes; (void)n_in; (void)out_size; (void)ws_size;
  auto F = [&](int i) { return (const float*)d_in[i]; };

  const float* x  = F(0);
  const int*   ei = (const int*)d_in[1];
  const float* ea = F(2);
  // params flattened in sorted-dict order (jax pytree):
  const float* bn_b[3] = {F(3), F(5), F(7)};
  const float* bn_g[3] = {F(4), F(6), F(8)};
  struct ConvP { const float *W, *We, *att_dst, *att_edge, *att_src, *bias; };
  ConvP conv[3];
  for (int l = 0; l < 3; ++l) {
    int b0 = 9 + 6 * l;
    conv[l] = {F(b0), F(b0 + 1), F(b0 + 2), F(b0 + 3), F(b0 + 4), F(b0 + 5)};
  }
  const float *em1b = F(27), *em1w = F(28), *em2b = F(29), *em2w = F(30),
              *em3b = F(31), *em3w = F(32);
  const float* lin_b[3] = {F(33), F(35), F(37)};
  const float* lin_w[3] = {F(34), F(36), F(38)};
  const float* ln_b[3]  = {F(39), F(41), F(43)};
  const float* ln_g[3]  = {F(40), F(42), F(44)};
  const float *out_b = F(45), *out_w = F(46);
  const float* res_b[3] = {F(47), F(49), F(51)};
  const float* res_w[3] = {F(48), F(50), F(52)};

  // ---- workspace carve (floats, 256B aligned) ----
  float* Wsp = (float*)d_ws;
  size_t off = 0;
  auto alloc = [&](size_t nf) { float* p = Wsp + off; off += (nf + 63) & ~(size_t)63; return p; };
  float*    xp   = alloc((size_t)N_NODES * 2048);
  float*    agg  = alloc((size_t)N_NODES * 2048);
  float*    e_f  = alloc((size_t)EF_EDGES * EDIM);
  float*    esum = alloc((size_t)N_NODES * EDIM);
  float*    cnt  = alloc(N_NODES);
  float*    ve   = alloc(128);
  float*    a_s  = alloc(N_NODES * HEADS);
  float*    a_d  = alloc(N_NODES * HEADS);
  float*    wls  = alloc((size_t)EF_EDGES * HEADS);
  unsigned* mEnc = (unsigned*)alloc(N_NODES * HEADS);
  float*    den  = alloc(N_NODES * HEADS);
  float*    stats= alloc(4096);
  float*    hl   = alloc((size_t)N_NODES * 256);
  float*    resv = alloc((size_t)N_NODES * 256);
  float*    hA   = alloc((size_t)N_NODES * 256);
  float*    hB   = alloc((size_t)N_NODES * 256);

  // ---- edge MLP + self-loop mean features ----
  edge_mlp_kernel<<<(E_EDGES + 255) / 256, 256, 0, stream>>>(
      ea, em1w, em1b, em2w, em2b, em3w, em3b, e_f);
  fill_f32<<<512, 256, 0, stream>>>(esum, 0.f, (long long)N_NODES * EDIM);
  fill_f32<<<64, 256, 0, stream>>>(cnt, 0.f, N_NODES);
  edge_scatter_sum<<<(E_EDGES * EDIM + 255) / 256, 256, 0, stream>>>(ei, e_f, esum, cnt);
  selfloop_mean<<<(N_NODES * EDIM + 255) / 256, 256, 0, stream>>>(esum, cnt, e_f);

  const float* hcur = x;
  int Cin = FIN;
  float* hbuf[2] = {hA, hB};
  const int Cd[3] = {64, 128, 256};

  for (int l = 0; l < 3; ++l) {
    int C = Cd[l], HC = HEADS * C, Dl = C;
    // xp = hcur @ W^T   (WMMA fp32, NT=4 strips: N % 128 handled by grid.y)
    dim3 gx(N_NODES / 16, (HC + 127) / 128);
    wmma_gemm_nt<4><<<gx, 64, 0, stream>>>(hcur, conv[l].W, nullptr, xp, N_NODES, HC, Cin);
    // attention scores
    att_node_kernel<<<(N_NODES * HEADS) / 8, 256, 0, stream>>>(
        xp, conv[l].att_src, conv[l].att_dst, a_s, a_d, C);
    att_edge_vec_kernel<<<1, 128, 0, stream>>>(conv[l].We, conv[l].att_edge, ve, C);
    fill_u32<<<64, 256, 0, stream>>>(mEnc, ENC_NEG_INF, N_NODES * HEADS);
    fill_f32<<<64, 256, 0, stream>>>(den, 0.f, N_NODES * HEADS);
    int egrid = (EF_EDGES * HEADS + 255) / 256;
    logits_max_kernel<<<egrid, 256, 0, stream>>>(ei, e_f, ve, a_s, a_d, wls, mEnc);
    expw_kernel<<<egrid, 256, 0, stream>>>(ei, wls, mEnc, den);
    // alpha-weighted scatter-sum (starts from bias)
    agg_init_kernel<<<32768, 256, 0, stream>>>(agg, conv[l].bias, HC);
    aggregate_kernel<<<131072, 256, 0, stream>>>(ei, xp, wls, den, agg, C);
    // BatchNorm + ReLU in place
    fill_f32<<<32, 256, 0, stream>>>(stats, 0.f, 2 * HC);
    dim3 bg(HC / 256, 64);
    bn_partial_kernel<<<bg, 256, 0, stream>>>(agg, stats, HC);
    bn_apply_kernel<<<32768, 256, 0, stream>>>(agg, stats, bn_g[l], bn_b[l], HC);
    // lin + residual GEMMs (WMMA fp32)
    dim3 gl(N_NODES / 16, (Dl + 127) / 128);
    wmma_gemm_nt<4><<<gl, 64, 0, stream>>>(agg, lin_w[l], lin_b[l], hl, N_NODES, Dl, HC);
    wmma_gemm_nt<4><<<gl, 64, 0, stream>>>(hcur, res_w[l], res_b[l], resv, N_NODES, Dl, Cin);
    // LayerNorm + ReLU + residual
    ln_relu_res_kernel<<<N_NODES / 8, 256, 0, stream>>>(
        hl, resv, ln_g[l], ln_b[l], hbuf[l & 1], Dl);
    hcur = hbuf[l & 1];
    Cin = Dl;
  }

  // final classifier: [N,256] @ [16,256]^T + b   (single 16-wide tile -> NT=1)
  dim3 go(N_NODES / 16, 1);
  wmma_gemm_nt<1><<<go, 64, 0, stream>>>(hcur, out_w, out_b, (float*)d_out,
                                         N_NODES, NCLS, 256);
}